// MemoryEfficientSVDPlaneProjection_32478542692446
// MI455X (gfx1250) — compile-verified
//
#include <hip/hip_runtime.h>
#include <math.h>

// ---------------------------------------------------------------------------
// MemoryEfficientSVDPlaneProjection for MI455X (gfx1250, wave32)
//   K1: plane normalization (B*M = 256 threads)
//   K2: WMMA 16x16x4 f32 distance tiles -> per-plane masked sums (cnt, S1, S2)
//       deterministic two-stage reduction (block partials in ws)
//   K3: centroid/cov + analytic smallest-eigenvector of 3x3 symmetric cov
//   K4: per-point sequential plane projection, writes proj + displacement
// ---------------------------------------------------------------------------

#define THRESH   0.015f
#define EPSN     1e-06f
#define EPSCOV   1e-08f

typedef __attribute__((ext_vector_type(2))) float v2f;
typedef __attribute__((ext_vector_type(8))) float v8f;

constexpr int  Bsz = 16;
constexpr int  Nsz = 131072;
constexpr int  Msz = 16;
constexpr int  CH  = 16;                 // reduction blocks per batch
constexpr int  PPB = Nsz / CH;           // points per block  (8192)
constexpr int  TPW = PPB / (8 * 16);     // 16-pt tiles per wave (64)

// workspace layout (floats)
constexpr int PLS       = 8;                                   // plane struct stride
constexpr int OFF_PLN   = 0;                                   // [B][M][8]  nhat.xyz, off, valid
constexpr int OFF_PART  = Bsz * Msz * PLS;                     // [B][CH][M][10]
constexpr int OFF_REF   = OFF_PART + Bsz * CH * Msz * 10;      // [B][M][8]  nref.xyz, dref, ok

// -------------------------------- K1 ---------------------------------------
__global__ void k_normalize_planes(const float* __restrict__ planes,
                                   float* __restrict__ ws) {
    int t = threadIdx.x;                       // t = b*16 + m  (256 threads)
    const float* p = planes + t * 4;
    float nx = p[0], ny = p[1], nz = p[2], off = p[3];
    float nrm   = sqrtf(nx * nx + ny * ny + nz * nz);
    float valid = (nrm > EPSN) ? 1.0f : 0.0f;
    float inv   = 1.0f / fmaxf(nrm, EPSN);
    float* o = ws + OFF_PLN + t * PLS;
    o[0] = nx * inv; o[1] = ny * inv; o[2] = nz * inv;
    o[3] = off;      o[4] = valid;
    o[5] = 0.0f;     o[6] = 0.0f;     o[7] = 0.0f;
}

// -------------------------------- K2 ---------------------------------------
// Each wave: A = 16 points padded to (x,y,z,1); B = 16 planes (nx,ny,nz,off).
// v_wmma_f32_16x16x4_f32 produces the 16x16 signed-distance tile with the
// plane index in the lane and the point index in the accumulator VGPR --
// exactly the layout needed for per-plane (cnt, S1, S2) accumulation.
__global__ void __launch_bounds__(256)
k_reduce(const float* __restrict__ pts, float* __restrict__ ws) {
    const int b     = blockIdx.x / CH;
    const int chunk = blockIdx.x % CH;
    const int wave  = threadIdx.x >> 5;
    const int lane  = threadIdx.x & 31;
    const int pm    = lane & 15;               // plane owned by this lane

    const float* pl = ws + OFF_PLN + (b * Msz + pm) * PLS;
    v2f bmat;
    bmat.x = (lane < 16) ? pl[0] : pl[2];      // K=0 -> nx  | K=2 -> nz
    bmat.y = (lane < 16) ? pl[1] : pl[3];      // K=1 -> ny  | K=3 -> off
    const float valid = pl[4];

    float cnt = 0.f, sx = 0.f, sy = 0.f, sz = 0.f;
    float sxx = 0.f, sxy = 0.f, sxz = 0.f, syy = 0.f, syz = 0.f, szz = 0.f;

    const size_t chunkBase = (size_t)b * Nsz + (size_t)chunk * PPB;
    const int    pbase     = (lane >> 4) << 3; // accumulator half: points 0..7 / 8..15

    for (int i = 0; i < TPW; ++i) {
        const size_t pidx = chunkBase + (size_t)(i * 8 + wave) * 16 + (size_t)(lane & 15);
        const float* pp = pts + pidx * 3;
        float x = pp[0], y = pp[1], z = pp[2];
        __builtin_prefetch(pp + 3 * 128, 0, 1);        // next iteration's tile group

        v2f a;
        a.x = (lane < 16) ? x : z;                      // K=0 -> x | K=2 -> z
        a.y = (lane < 16) ? y : 1.0f;                   // K=1 -> y | K=3 -> 1
        v8f c = {0.f, 0.f, 0.f, 0.f, 0.f, 0.f, 0.f, 0.f};
        v8f d = __builtin_amdgcn_wmma_f32_16x16x4_f32(
            false, a, false, bmat, (short)0, c, false, false);

#pragma unroll
        for (int r = 0; r < 8; ++r) {
            float msk = ((fabsf(d[r]) < THRESH) && (valid > 0.5f)) ? 1.0f : 0.0f;
            int src = pbase + r;                        // lane src holds point src
            float px = __shfl(x, src, 32);
            float py = __shfl(y, src, 32);
            float pz = __shfl(z, src, 32);
            cnt += msk;
            sx  = fmaf(msk, px, sx);
            sy  = fmaf(msk, py, sy);
            sz  = fmaf(msk, pz, sz);
            sxx = fmaf(msk * px, px, sxx);
            sxy = fmaf(msk * px, py, sxy);
            sxz = fmaf(msk * px, pz, sxz);
            syy = fmaf(msk * py, py, syy);
            syz = fmaf(msk * py, pz, syz);
            szz = fmaf(msk * pz, pz, szz);
        }
    }

    // combine the two lane-halves of the same plane
    cnt += __shfl_xor(cnt, 16, 32);
    sx  += __shfl_xor(sx, 16, 32);  sy  += __shfl_xor(sy, 16, 32);
    sz  += __shfl_xor(sz, 16, 32);  sxx += __shfl_xor(sxx, 16, 32);
    sxy += __shfl_xor(sxy, 16, 32); sxz += __shfl_xor(sxz, 16, 32);
    syy += __shfl_xor(syy, 16, 32); syz += __shfl_xor(syz, 16, 32);
    szz += __shfl_xor(szz, 16, 32);

    __shared__ float lds[8][16][10];
    if (lane < 16) {
        float* o = lds[wave][pm];
        o[0] = cnt; o[1] = sx;  o[2] = sy;  o[3] = sz;  o[4] = sxx;
        o[5] = sxy; o[6] = sxz; o[7] = syy; o[8] = syz; o[9] = szz;
    }
    __syncthreads();

    int t = threadIdx.x;
    if (t < 160) {
        int m = t / 10, v = t % 10;
        float s = 0.f;
#pragma unroll
        for (int w = 0; w < 8; ++w) s += lds[w][m][v];
        ws[OFF_PART + (((size_t)b * CH + chunk) * Msz + m) * 10 + v] = s;
    }
}

// -------------------------------- K3 ---------------------------------------
__global__ void k_finalize(float* __restrict__ ws) {
    int t = threadIdx.x;                      // 256 = B*M
    int b = t >> 4, m = t & 15;

    float s[10];
#pragma unroll
    for (int v = 0; v < 10; ++v) s[v] = 0.f;
    for (int c = 0; c < CH; ++c) {
        const float* p = ws + OFF_PART + (((size_t)b * CH + c) * Msz + m) * 10;
#pragma unroll
        for (int v = 0; v < 10; ++v) s[v] += p[v];
    }

    float cnt   = s[0];
    float denom = fmaxf(cnt, 1.0f);
    float cx = s[1] / denom, cy = s[2] / denom, cz = s[3] / denom;

    // cov = (S2 - c*S1^T - S1*c^T + cnt*c*c^T)/denom + eps*I
    float a00 = (s[4] - 2.f * cx * s[1] + cnt * cx * cx) / denom + EPSCOV;
    float a01 = (s[5] - cx * s[2] - cy * s[1] + cnt * cx * cy) / denom;
    float a02 = (s[6] - cx * s[3] - cz * s[1] + cnt * cx * cz) / denom;
    float a11 = (s[7] - 2.f * cy * s[2] + cnt * cy * cy) / denom + EPSCOV;
    float a12 = (s[8] - cy * s[3] - cz * s[2] + cnt * cy * cz) / denom;
    float a22 = (s[9] - 2.f * cz * s[3] + cnt * cz * cz) / denom + EPSCOV;

    const float* pl = ws + OFF_PLN + (b * Msz + m) * PLS;
    float fx = pl[0], fy = pl[1], fz = pl[2];      // fallback / sign reference

    // smallest eigenvector of symmetric 3x3 (trig eigenvalues + row crosses)
    float nx = fx, ny = fy, nz = fz;
    float p1 = a01 * a01 + a02 * a02 + a12 * a12;
    float q  = (a00 + a11 + a22) * (1.0f / 3.0f);
    float b00 = a00 - q, b11 = a11 - q, b22 = a22 - q;
    float p2 = b00 * b00 + b11 * b11 + b22 * b22 + 2.0f * p1;
    float pr = sqrtf(fmaxf(p2, 0.0f) * (1.0f / 6.0f));
    if (pr > 1e-12f) {
        float inv = 1.0f / pr;
        float d00 = b00 * inv, d11 = b11 * inv, d22 = b22 * inv;
        float d01 = a01 * inv, d02 = a02 * inv, d12 = a12 * inv;
        float detB = d00 * (d11 * d22 - d12 * d12)
                   - d01 * (d01 * d22 - d12 * d02)
                   + d02 * (d01 * d12 - d11 * d02);
        float r   = fminf(1.0f, fmaxf(-1.0f, 0.5f * detB));
        float phi = acosf(r) * (1.0f / 3.0f);
        float lmin = q + 2.0f * pr * cosf(phi + 2.09439510239f);   // +2*pi/3

        float m00 = a00 - lmin, m11 = a11 - lmin, m22 = a22 - lmin;
        // cross products of rows of (A - lmin*I)
        float c1x = a01 * a12 - a02 * m11, c1y = a02 * a01 - m00 * a12, c1z = m00 * m11 - a01 * a01;
        float c2x = a01 * m22 - a02 * a12, c2y = a02 * a02 - m00 * m22, c2z = m00 * a12 - a01 * a02;
        float c3x = m11 * m22 - a12 * a12, c3y = a12 * a02 - a01 * m22, c3z = a01 * a12 - m11 * a02;
        float n1 = c1x * c1x + c1y * c1y + c1z * c1z;
        float n2 = c2x * c2x + c2y * c2y + c2z * c2z;
        float n3 = c3x * c3x + c3y * c3y + c3z * c3z;
        float bx = c1x, by = c1y, bz = c1z, bn = n1;
        if (n2 > bn) { bx = c2x; by = c2y; bz = c2z; bn = n2; }
        if (n3 > bn) { bx = c3x; by = c3y; bz = c3z; bn = n3; }
        if (bn > 1e-24f) {
            float rin = rsqrtf(bn);
            nx = bx * rin; ny = by * rin; nz = bz * rin;
        }
    }

    // sign alignment with input normal
    float dsign = nx * fx + ny * fy + nz * fz;
    if (dsign < 0.0f) { nx = -nx; ny = -ny; nz = -nz; }
    float dref = -(cx * nx + cy * ny + cz * nz);
    float ok   = (cnt >= 3.0f) ? 1.0f : 0.0f;

    float* o = ws + OFF_REF + (b * Msz + m) * PLS;
    o[0] = nx; o[1] = ny; o[2] = nz; o[3] = dref; o[4] = ok;
    o[5] = 0.f; o[6] = 0.f; o[7] = 0.f;
}

// -------------------------------- K4 ---------------------------------------
__global__ void __launch_bounds__(256)
k_project(const float* __restrict__ pts, const float* __restrict__ ws,
          float* __restrict__ out) {
    constexpr int BPB = Nsz / 256;            // blocks per batch (512)
    const int b = blockIdx.x / BPB;
    const size_t n = (size_t)(blockIdx.x % BPB) * 256 + threadIdx.x;

    __shared__ float sN[16][4];               // normalized plane + offset
    __shared__ float sVal[16];
    __shared__ float sR[16][4];               // refined plane (nref, dref)
    __shared__ float sOk[16];
    if (threadIdx.x < 16) {
        int m = threadIdx.x;
        const float* pl = ws + OFF_PLN + (b * Msz + m) * PLS;
        sN[m][0] = pl[0]; sN[m][1] = pl[1]; sN[m][2] = pl[2]; sN[m][3] = pl[3];
        sVal[m]  = pl[4];
        const float* rf = ws + OFF_REF + (b * Msz + m) * PLS;
        sR[m][0] = rf[0]; sR[m][1] = rf[1]; sR[m][2] = rf[2]; sR[m][3] = rf[3];
        sOk[m]   = rf[4];
    }
    __syncthreads();

    const size_t g  = (size_t)b * Nsz + n;
    const float* pp = pts + g * 3;
    float px = pp[0], py = pp[1], pz = pp[2];

    unsigned cm = 0;
#pragma unroll
    for (int m = 0; m < 16; ++m) {
        float d = fabsf(px * sN[m][0] + py * sN[m][1] + pz * sN[m][2] + sN[m][3]);
        if ((d < THRESH) && (sVal[m] > 0.5f)) cm |= (1u << m);
    }

    float qx = px, qy = py, qz = pz;
#pragma unroll
    for (int m = 0; m < 16; ++m) {
        float u = ((cm >> m) & 1u) ? sOk[m] : 0.0f;     // upd = mask * ok
        float t = qx * sR[m][0] + qy * sR[m][1] + qz * sR[m][2] + sR[m][3];
        float sft = u * t;
        qx = fmaf(-sft, sR[m][0], qx);
        qy = fmaf(-sft, sR[m][1], qy);
        qz = fmaf(-sft, sR[m][2], qz);
    }

    float* o1 = out + g * 3;
    o1[0] = qx; o1[1] = qy; o1[2] = qz;
    float* o2 = out + (size_t)Bsz * Nsz * 3 + g * 3;
    o2[0] = qx - px; o2[1] = qy - py; o2[2] = qz - pz;
}

// ---------------------------------------------------------------------------
extern "C" void kernel_launch(void* const* d_in, const int* in_sizes, int n_in,
                              void* d_out, int out_size, void* d_ws, size_t ws_size,
                              hipStream_t stream) {
    const float* pts    = (const float*)d_in[0];   // (16, 131072, 3) f32
    const float* planes = (const float*)d_in[1];   // (16, 16, 4)     f32
    float* ws  = (float*)d_ws;                     // needs ~176 KB
    float* out = (float*)d_out;                    // proj || displacement

    hipLaunchKernelGGL(k_normalize_planes, dim3(1),              dim3(256), 0, stream, planes, ws);
    hipLaunchKernelGGL(k_reduce,           dim3(Bsz * CH),       dim3(256), 0, stream, pts, ws);
    hipLaunchKernelGGL(k_finalize,         dim3(1),              dim3(256), 0, stream, ws);
    hipLaunchKernelGGL(k_project,          dim3(Bsz * (Nsz/256)),dim3(256), 0, stream, pts, ws, out);
}